// NeuroSAT_3023656977003
// MI455X (gfx1250) — compile-verified
//
#include <hip/hip_runtime.h>
#include <math.h>

// NeuroSAT forward for gfx1250 (MI455X). fp32 WMMA (V_WMMA_F32_16X16X4_F32)
// for all dense GEMMs with LDS-staged B panels; deterministic CSR segment
// sums; device-side early-exit flag so the 26-round loop is graph-capturable.

#define D      128
#define LITS   2048
#define NVARS  1024
#define CLS    4096
#define NE     16384
#define N_ROUNDS 26

typedef __attribute__((ext_vector_type(2))) float v2f;
typedef __attribute__((ext_vector_type(8))) float v8f;

__device__ __forceinline__ float sigm_(float x) { return 1.f / (1.f + expf(-x)); }

// ---------------------------------------------------------------------------
// GEMM: C[M x NCOLS] = act( A[M x KDIM] @ B[KDIM x NCOLS] + bias (+C if ACC) )
//       (* rowscale).  Row-major everywhere.
// Block = 8 waves; all waves share one 16*NT-wide column strip (B^T panel
// staged once in LDS, bank-conflict-free with K+4 padding) and cover 8
// consecutive 16-row M tiles.  Inner loop per k-step:
//   1x global_load_b64 (A frag) + NT x ds_load_b64 (B frag) + NT x WMMA.
// f32 WMMA fragment layouts per CDNA5 ISA 7.12.2:
//   A 16x4 : lane L (lr=L&15, half=L>>4): v0=A[lr][k+2h], v1=A[lr][k+2h+1]
//   B 4x16 : lane L: v0=B[k+2h][lr], v1=B[k+2h+1][lr]
//   C 16x16: vgpr r holds M=r+8h, N=lr
// ---------------------------------------------------------------------------
template <int NT, int NCOLS, int KDIM, int ACT, bool ACC, bool SCALE>
__global__ __launch_bounds__(256) void gemm_f32_wmma(
    const float* __restrict__ A, int lda,
    const float* __restrict__ B,
    const float* __restrict__ bias,
    const float* __restrict__ rowscale,
    float* __restrict__ C, int ldc, int M)
{
  constexpr int STRIDE = KDIM + 4;              // (K+4)%64==4 -> conflict-free
  constexpr int STRIPS = NCOLS / (16 * NT);
  __shared__ float bt[16 * NT * STRIDE];        // B^T panel: [local col][k]

  const int strip  = blockIdx.x % STRIPS;
  const int mblock = blockIdx.x / STRIPS;
  const int col0   = strip * 16 * NT;

  // Cooperative transposed load of the B panel into LDS.
  {
    const int cc = threadIdx.x & 15;            // float4 column group
    for (int kk = threadIdx.x >> 4; kk < KDIM; kk += 16) {
      float4 v = *(const float4*)(B + (size_t)kk * NCOLS + col0 + 4 * cc);
      bt[(4 * cc + 0) * STRIDE + kk] = v.x;
      bt[(4 * cc + 1) * STRIDE + kk] = v.y;
      bt[(4 * cc + 2) * STRIDE + kk] = v.z;
      bt[(4 * cc + 3) * STRIDE + kk] = v.w;
    }
  }
  __syncthreads();

  const int tileM = mblock * 8 + (threadIdx.x >> 5);
  const int row0  = tileM * 16;
  const int lane  = threadIdx.x & 31;
  const int half  = lane >> 4;
  const int lr    = lane & 15;

  v8f acc[NT];
#pragma unroll
  for (int t = 0; t < NT; ++t)
#pragma unroll
    for (int r = 0; r < 8; ++r) acc[t][r] = 0.f;

  const float* Arow = A + (size_t)(row0 + lr) * lda + 2 * half;
  const float* btl  = bt + lr * STRIDE + 2 * half;
  __builtin_prefetch(Arow, 0, 3);               // global_prefetch_b8

#pragma unroll 4
  for (int k = 0; k < KDIM; k += 4) {
    v2f a;
    a.x = Arow[k];
    a.y = Arow[k + 1];
#pragma unroll
    for (int t = 0; t < NT; ++t) {
      v2f b;
      b.x = btl[16 * t * STRIDE + k];           // ds_load_b64 pair
      b.y = btl[16 * t * STRIDE + k + 1];
      acc[t] = __builtin_amdgcn_wmma_f32_16x16x4_f32(
          false, a, false, b, (short)0, acc[t], false, false);
    }
  }

#pragma unroll
  for (int t = 0; t < NT; ++t) {
    int c  = col0 + 16 * t + lr;
    float bv = bias ? bias[c] : 0.f;
#pragma unroll
    for (int r = 0; r < 8; ++r) {
      int m = row0 + r + 8 * half;
      float v = acc[t][r] + bv;
      if (ACC) v += C[(size_t)m * ldc + c];
      if (ACT == 1) v = fmaxf(v, 0.f);
      if (SCALE) v *= rowscale[m];
      C[(size_t)m * ldc + c] = v;
    }
  }
}

// X[e] = [ Ha[ia[e]] , Hb[ib[e]] ]  (E x 2D), block = 64 threads (float4 each)
__global__ void k_concat_gather(const float* __restrict__ Ha, const int* __restrict__ ia,
                                const float* __restrict__ Hb, const int* __restrict__ ib,
                                float* __restrict__ X)
{
  int e = blockIdx.x, t = threadIdx.x;
  float4* dst = (float4*)(X + (size_t)e * 2 * D);
  if (t < 32) {
    dst[t] = ((const float4*)(Ha + (size_t)ia[e] * D))[t];
  } else {
    dst[t] = ((const float4*)(Hb + (size_t)ib[e] * D))[t - 32];
  }
}

// X[l] = [ agg_c[l], lit_h[flip(l)] ]  (LITS x 2D)
__global__ void k_concat_flip(const float* __restrict__ aggc,
                              const float* __restrict__ lith,
                              float* __restrict__ X)
{
  int l = blockIdx.x, t = threadIdx.x;
  float4* dst = (float4*)(X + (size_t)l * 2 * D);
  if (t < 32) {
    dst[t] = ((const float4*)(aggc + (size_t)l * D))[t];
  } else {
    int fl = (l < NVARS) ? l + NVARS : l - NVARS;
    dst[t] = ((const float4*)(lith + (size_t)fl * D))[t - 32];
  }
}

// Deterministic CSR build (serial, once per launch; ~50K scalar iters)
__global__ void k_build_csr(const int* __restrict__ key, int E, int nseg,
                            int* __restrict__ off, int* __restrict__ cur,
                            int* __restrict__ lst)
{
  if (blockIdx.x == 0 && threadIdx.x == 0) {
    for (int s = 0; s <= nseg; ++s) off[s] = 0;
    for (int e = 0; e < E; ++e) off[key[e] + 1]++;
    for (int s = 0; s < nseg; ++s) off[s + 1] += off[s];
    for (int s = 0; s < nseg; ++s) cur[s] = off[s];
    for (int e = 0; e < E; ++e) { int k = key[e]; lst[cur[k]++] = e; }
  }
}

// out[s][:] = sum over CSR edge list of Msrc[e][:], deterministic order
__global__ void k_segsum(const float* __restrict__ Msrc, const int* __restrict__ off,
                         const int* __restrict__ lst, float* __restrict__ out)
{
  int s = blockIdx.x, j = threadIdx.x;  // 128 threads
  float acc = 0.f;
  int b = off[s], e2 = off[s + 1];
  for (int p = b; p < e2; ++p) acc += Msrc[(size_t)lst[p] * D + j];
  out[(size_t)s * D + j] = acc;
}

// out[row] = (sigmoid?)( dot(H[row,:], w2) + b2 ), one wave per row
template <bool SIG>
__global__ void k_rowdot(const float* __restrict__ H, const float* __restrict__ w2,
                         const float* __restrict__ b2, float* __restrict__ out, int Mrows)
{
  int wave = blockIdx.x * (blockDim.x >> 5) + (threadIdx.x >> 5);
  if (wave >= Mrows) return;
  int lane = threadIdx.x & 31;
  const float* row = H + (size_t)wave * D;
  float s = 0.f;
  for (int j = lane; j < D; j += 32) s += row[j] * w2[j];
  for (int o = 16; o > 0; o >>= 1) s += __shfl_xor(s, o, 32);
  if (lane == 0) {
    float v = s + b2[0];
    out[wave] = SIG ? (1.f / (1.f + expf(-v))) : v;
  }
}

// LSTM gates: G[row] = [i f g o] (4*D); c_out = sig(f)*c + sig(i)*tanh(g);
// z_out = h_in + sig(o)*tanh(c_out)   (residual for BN)
__global__ void k_lstm_gates(const float* __restrict__ G, const float* __restrict__ h_in,
                             const float* __restrict__ c_in,
                             float* __restrict__ c_out, float* __restrict__ z_out)
{
  int r = blockIdx.x, j = threadIdx.x;  // 128 threads
  const float* g = G + (size_t)r * 4 * D;
  float vi = g[j], vf = g[D + j], vg = g[2 * D + j], vo = g[3 * D + j];
  float c2 = sigm_(vf) * c_in[(size_t)r * D + j] + sigm_(vi) * tanhf(vg);
  float hn = sigm_(vo) * tanhf(c2);
  c_out[(size_t)r * D + j] = c2;
  z_out[(size_t)r * D + j] = h_in[(size_t)r * D + j] + hn;
}

// Per-column mean / rsqrt(var + 1e-5) over `rows` rows
__global__ void k_bn_stats(const float* __restrict__ Z, int rows,
                           float* __restrict__ mu, float* __restrict__ rs)
{
  int j = blockIdx.x, t = threadIdx.x;  // 256 threads
  float s = 0.f, s2 = 0.f;
  for (int r = t; r < rows; r += 256) {
    float v = Z[(size_t)r * D + j];
    s += v; s2 += v * v;
  }
  __shared__ float sh[256], sh2[256];
  sh[t] = s; sh2[t] = s2; __syncthreads();
  for (int o = 128; o > 0; o >>= 1) {
    if (t < o) { sh[t] += sh[t + o]; sh2[t] += sh2[t + o]; }
    __syncthreads();
  }
  if (t == 0) {
    float m = sh[0] / rows;
    float var = sh2[0] / rows - m * m;
    mu[j] = m;
    rs[j] = rsqrtf(var + 1e-5f);
  }
}

__global__ void k_bn_apply(const float* __restrict__ Z, const float* __restrict__ mu,
                           const float* __restrict__ rs, const float* __restrict__ g,
                           const float* __restrict__ b, float* __restrict__ out, int rows)
{
  int idx = blockIdx.x * blockDim.x + threadIdx.x;
  if (idx >= rows * D) return;
  int j = idx & (D - 1);
  out[idx] = (Z[idx] - mu[j]) * rs[j] * g[j] + b[j];
}

// Partial sums of ||new-old||^2 and ||old||^2 (fixed 64-block grid)
__global__ void k_norm_partial(const float* __restrict__ lh_new,
                               const float* __restrict__ lh_old,
                               float* __restrict__ part)
{
  int t = threadIdx.x;
  const int n = LITS * D;
  float d2 = 0.f, p2 = 0.f;
  for (int i = blockIdx.x * blockDim.x + t; i < n; i += gridDim.x * blockDim.x) {
    float a = lh_new[i] - lh_old[i];
    d2 += a * a;
    float b = lh_old[i];
    p2 += b * b;
  }
  __shared__ float sd[256], sp[256];
  sd[t] = d2; sp[t] = p2; __syncthreads();
  for (int o = 128; o > 0; o >>= 1) {
    if (t < o) { sd[t] += sd[t + o]; sp[t] += sp[t + o]; }
    __syncthreads();
  }
  if (t == 0) { part[blockIdx.x] = sd[0]; part[gridDim.x + blockIdx.x] = sp[0]; }
}

// Commit new state unless already converged (uses done flag from *before* update)
__global__ void k_commit(float* __restrict__ lit_h, float* __restrict__ lit_c,
                         float* __restrict__ cls_h, float* __restrict__ cls_c,
                         const float* __restrict__ lh, const float* __restrict__ lc,
                         const float* __restrict__ ch, const float* __restrict__ cc,
                         const int* __restrict__ done)
{
  if (*done) return;
  int idx = blockIdx.x * blockDim.x + threadIdx.x;
  if (idx < LITS * D) { lit_h[idx] = lh[idx]; lit_c[idx] = lc[idx]; }
  if (idx < CLS * D)  { cls_h[idx] = ch[idx]; cls_c[idx] = cc[idx]; }
}

__global__ void k_update_done(const float* __restrict__ part, int npart,
                              int* __restrict__ done)
{
  if (threadIdx.x == 0) {
    float d2 = 0.f, p2 = 0.f;
    for (int i = 0; i < npart; ++i) { d2 += part[i]; p2 += part[npart + i]; }
    float change = sqrtf(d2) / (sqrtf(p2) + 1e-6f);
    if (change < 0.01f) *done = 1;
  }
}

__global__ void k_init(const float* __restrict__ liW, const float* __restrict__ lib,
                       const float* __restrict__ clW, const float* __restrict__ clb,
                       float* __restrict__ lit_h, float* __restrict__ lit_c,
                       float* __restrict__ cls_h, float* __restrict__ cls_c,
                       int* __restrict__ done)
{
  int idx = blockIdx.x * blockDim.x + threadIdx.x;
  int j = idx & (D - 1);
  if (idx < LITS * D) { lit_h[idx] = liW[j] + lib[j]; lit_c[idx] = 0.f; }
  if (idx < CLS * D)  { cls_h[idx] = clW[j] + clb[j]; cls_c[idx] = 0.f; }
  if (idx == 0) *done = 0;
}

__global__ void k_final(const float* __restrict__ v, float* __restrict__ out)
{
  int t = threadIdx.x;  // 256
  float s = 0.f;
  for (int i = t; i < NVARS; i += 256) {
    float pos = v[i], neg = v[i + NVARS];
    s += (fabsf(pos) > fabsf(neg)) ? pos : -neg;
  }
  __shared__ float sh[256];
  sh[t] = s; __syncthreads();
  for (int o = 128; o > 0; o >>= 1) {
    if (t < o) sh[t] += sh[t + o];
    __syncthreads();
  }
  if (t == 0) {
    float m = sh[0] / (float)NVARS;
    out[0] = 1.f / (1.f + expf(-m));
  }
}

// ---------------------------------------------------------------------------
// Host driver. Input order = setup_inputs() insertion order (params dict
// leaves first, then edge_src, edge_dst, n_vars, n_clauses).
// ---------------------------------------------------------------------------
#define GEMM(NT_, N_, K_, ACT_, ACC_, SC_, Ap, lda_, Bp, biasp, scp, Cp, ldc_, M_)  \
  do {                                                                              \
    int blocks_ = ((M_) / 128) * ((N_) / (16 * (NT_)));                             \
    gemm_f32_wmma<NT_, N_, K_, ACT_, ACC_, SC_><<<blocks_, 256, 0, stream>>>(       \
        Ap, lda_, Bp, biasp, scp, Cp, ldc_, M_);                                    \
  } while (0)

extern "C" void kernel_launch(void* const* d_in, const int* in_sizes, int n_in,
                              void* d_out, int out_size, void* d_ws, size_t ws_size,
                              hipStream_t stream)
{
  (void)in_sizes; (void)n_in; (void)out_size; (void)ws_size;
  const float* p[42];
  for (int i = 0; i < 42; ++i) p[i] = (const float*)d_in[i];
  const int* edge_src = (const int*)d_in[42];
  const int* edge_dst = (const int*)d_in[43];

  float* base = (float*)d_ws;
  size_t off = 0;
  auto F = [&](size_t n) { float* q = base + off; off += n; return q; };
  float* lit_h  = F((size_t)LITS * D);
  float* lit_c  = F((size_t)LITS * D);
  float* cls_h  = F((size_t)CLS * D);
  float* cls_c  = F((size_t)CLS * D);
  float* lh_tmp = F((size_t)LITS * D);
  float* lc_tmp = F((size_t)LITS * D);
  float* ch_tmp = F((size_t)CLS * D);
  float* cc_tmp = F((size_t)CLS * D);
  float* X      = F((size_t)NE * 2 * D);
  float* bufA   = F((size_t)NE * D);
  float* bufB   = F((size_t)NE * D);
  float* wv     = F((size_t)NE);
  float* agg_l  = F((size_t)CLS * D);
  float* agg_c  = F((size_t)LITS * D);
  float* G      = F((size_t)CLS * 4 * D);
  float* zbuf   = F((size_t)CLS * D);
  float* mu     = F(D);
  float* rs     = F(D);
  float* part   = F(128);
  float* votes  = F(LITS);
  int* ibase = (int*)(base + off);
  size_t io = 0;
  auto I = [&](size_t n) { int* q = ibase + io; io += n; return q; };
  int* done = I(4);        // [0] used; padded
  int* offC = I(CLS + 1);
  int* curC = I(CLS);
  int* lstC = I(NE);
  int* offL = I(LITS + 1);
  int* curL = I(LITS);
  int* lstL = I(NE);

  k_build_csr<<<1, 1, 0, stream>>>(edge_dst, NE, CLS, offC, curC, lstC);
  k_build_csr<<<1, 1, 0, stream>>>(edge_src, NE, LITS, offL, curL, lstL);

  k_init<<<(CLS * D + 255) / 256, 256, 0, stream>>>(p[0], p[1], p[2], p[3],
                                                    lit_h, lit_c, cls_h, cls_c, done);

  for (int round = 0; round < N_ROUNDS; ++round) {
    // --- literal -> clause messages -------------------------------------
    k_concat_gather<<<NE, 64, 0, stream>>>(lit_h, edge_src, cls_h, edge_dst, X);
    GEMM(4, D, 2 * D, 1, false, false, X, 2 * D, p[4], p[5], nullptr, bufA, D, NE);   // lit_attn h
    k_rowdot<true><<<NE / 8, 256, 0, stream>>>(bufA, p[6], p[7], wv, NE);             // lit_attn w
    GEMM(4, D, D, 1, false, false, X, 2 * D, p[12], p[13], nullptr, bufA, D, NE);     // msg t1 (src_h)
    GEMM(4, D, D, 1, false, false, bufA, D, p[14], p[15], nullptr, bufB, D, NE);      // msg t2
    GEMM(4, D, D, 0, false, true,  bufB, D, p[16], p[17], wv, bufA, D, NE);           // m = t3 * w
    k_segsum<<<CLS, D, 0, stream>>>(bufA, offC, lstC, agg_l);
    GEMM(4, 4 * D, D, 0, false, false, agg_l, D, p[28], p[30], nullptr, G, 4 * D, CLS); // Wih+bih
    GEMM(4, 4 * D, D, 0, true,  false, cls_h, D, p[29], p[31], nullptr, G, 4 * D, CLS); // +Whh+bhh
    k_lstm_gates<<<CLS, D, 0, stream>>>(G, cls_h, cls_c, cc_tmp, zbuf);
    k_bn_stats<<<D, 256, 0, stream>>>(zbuf, CLS, mu, rs);
    k_bn_apply<<<(CLS * D + 255) / 256, 256, 0, stream>>>(zbuf, mu, rs, p[34], p[35], ch_tmp, CLS);

    // --- clause -> literal messages -------------------------------------
    k_concat_gather<<<NE, 64, 0, stream>>>(ch_tmp, edge_dst, lit_h, edge_src, X);
    GEMM(4, D, 2 * D, 1, false, false, X, 2 * D, p[8], p[9], nullptr, bufA, D, NE);   // clause_attn h
    k_rowdot<true><<<NE / 8, 256, 0, stream>>>(bufA, p[10], p[11], wv, NE);
    GEMM(4, D, D, 1, false, false, X, 2 * D, p[18], p[19], nullptr, bufA, D, NE);     // msg t1 (csrc)
    GEMM(4, D, D, 1, false, false, bufA, D, p[20], p[21], nullptr, bufB, D, NE);
    GEMM(4, D, D, 0, false, true,  bufB, D, p[22], p[23], wv, bufA, D, NE);
    k_segsum<<<LITS, D, 0, stream>>>(bufA, offL, lstL, agg_c);
    k_concat_flip<<<LITS, 64, 0, stream>>>(agg_c, lit_h, X);
    GEMM(4, 4 * D, 2 * D, 0, false, false, X, 2 * D, p[24], p[26], nullptr, G, 4 * D, LITS);
    GEMM(4, 4 * D, D, 0, true,  false, lit_h, D, p[25], p[27], nullptr, G, 4 * D, LITS);
    k_lstm_gates<<<LITS, D, 0, stream>>>(G, lit_h, lit_c, lc_tmp, zbuf);
    k_bn_stats<<<D, 256, 0, stream>>>(zbuf, LITS, mu, rs);
    k_bn_apply<<<(LITS * D + 255) / 256, 256, 0, stream>>>(zbuf, mu, rs, p[32], p[33], lh_tmp, LITS);

    // --- convergence + gated commit -------------------------------------
    k_norm_partial<<<64, 256, 0, stream>>>(lh_tmp, lit_h, part);
    k_commit<<<(CLS * D + 255) / 256, 256, 0, stream>>>(lit_h, lit_c, cls_h, cls_c,
                                                        lh_tmp, lc_tmp, ch_tmp, cc_tmp, done);
    k_update_done<<<1, 1, 0, stream>>>(part, 64, done);
  }

  // --- vote head ---------------------------------------------------------
  GEMM(4, D, D, 1, false, false, lit_h, D, p[36], p[37], nullptr, bufA, D, LITS);
  GEMM(4, D, D, 1, false, false, bufA, D, p[38], p[39], nullptr, bufB, D, LITS);
  k_rowdot<false><<<LITS / 8, 256, 0, stream>>>(bufB, p[40], p[41], votes, LITS);
  k_final<<<1, 256, 0, stream>>>(votes, (float*)d_out);
}